// EffiGlobalPointer_78271484003189
// MI455X (gfx1250) — compile-verified
//
#include <hip/hip_runtime.h>
#include <cstdint>

typedef __attribute__((ext_vector_type(2))) float v2f;
typedef __attribute__((ext_vector_type(8))) float v8f;

#define BB   8
#define LL   1024
#define HH   1024
#define DD   64
#define NENT 9
#define NCOL 160        // 128 (W1) + 18 (W2) padded to 160
#define NEPS 1.0e12f

// ---------------------------------------------------------------------------
// Kernel 0: pack [W1 | W2 | 0] transposed into Wct[NCOL][HH] so the GEMM
// B-fragment load is a single contiguous v2f per lane with no branching.
// ---------------------------------------------------------------------------
__global__ __launch_bounds__(256) void prep_kernel(
    const float* __restrict__ W1, const float* __restrict__ W2,
    float* __restrict__ Wct)
{
    const int t = blockIdx.x * 256 + threadIdx.x;   // 0 .. NCOL*HH-1
    const int n = t >> 10;                          // output column 0..159
    const int k = t & (HH - 1);                     // reduction index
    float v = 0.0f;
    if (n < 128)      v = W1[(size_t)k * 128 + n];
    else if (n < 146) v = W2[(size_t)k * 18 + (n - 128)];
    Wct[t] = v;                                     // coalesced in k
}

// ---------------------------------------------------------------------------
// Kernel 1: projection GEMM  proj = lhs @ [W1|W2]  via V_WMMA_F32_16X16X4_F32.
// One wave per 16x16 tile; inner loop = 2 contiguous b64 loads + 1 WMMA.
// Epilogue scatters: n<128 even->q, odd->k (+b1); n in [128,146) even->col
// bias, odd->row bias ((v+b2)*0.5).
// ---------------------------------------------------------------------------
__global__ __launch_bounds__(256) void proj_kernel(
    const float* __restrict__ lhs, const float* __restrict__ Wct,
    const float* __restrict__ b1,  const float* __restrict__ b2,
    float* __restrict__ qraw, float* __restrict__ kraw,
    float* __restrict__ rb,   float* __restrict__ cb)
{
    const int wave = threadIdx.x >> 5;
    const int lane = threadIdx.x & 31;
    const int li   = lane & 15;
    const int hi   = lane >> 4;
    const int gw   = blockIdx.x * 8 + wave;   // 0..5119
    const int mt   = gw / (NCOL / 16);
    const int nt   = gw % (NCOL / 16);
    const int m0   = mt * 16;
    const int n    = nt * 16 + li;

    const float* arow = lhs + (size_t)(m0 + li) * HH + 2 * hi;
    const float* brow = Wct + (size_t)n * HH + 2 * hi;

    v8f acc = {};
    #pragma unroll 8
    for (int k0 = 0; k0 < HH; k0 += 4) {
        // A frag 16x4: lane<16 holds K=k0,k0+1; lane>=16 holds K=k0+2,k0+3.
        v2f a = *(const v2f*)(arow + k0);
        // B frag 4x16 mirrored: Wct is column-major of [W1|W2] -> contiguous.
        v2f bf = *(const v2f*)(brow + k0);
        acc = __builtin_amdgcn_wmma_f32_16x16x4_f32(
                  false, a, false, bf, (short)0, acc, false, false);
    }

    // C/D layout: lane holds (M = r + 8*hi, N = n) in acc[r]
    #pragma unroll
    for (int r = 0; r < 8; ++r) {
        const int grow = m0 + hi * 8 + r;        // global token row 0..8191
        const int bidx = grow >> 10;
        const int l    = grow & (LL - 1);
        float val = acc[r];
        if (n < 128) {
            val += b1[n];
            float* dst = (n & 1) ? kraw : qraw;
            dst[((size_t)grow << 6) + (n >> 1)] = val;
        } else if (n < 146) {
            const int c = n - 128;
            val = (val + b2[c]) * 0.5f;
            float* dst = (c & 1) ? rb : cb;      // odd channels add along rows
            dst[((size_t)(bidx * NENT + (c >> 1)) << 10) + l] = val;
        }
    }
}

// ---------------------------------------------------------------------------
// Kernel 2: in-place RoPE on q and k. One thread per (token, pair).
// ---------------------------------------------------------------------------
__global__ __launch_bounds__(256) void rope_kernel(
    float* __restrict__ q, float* __restrict__ k)
{
    const int t = blockIdx.x * blockDim.x + threadIdx.x;  // 0..B*L*32-1
    const int g = t >> 5;                                 // token index
    const int i = t & 31;                                 // pair index
    const int l = g & (LL - 1);                           // position
    // inv[i] = 10000^(-2i/64) = exp(-i * ln(10000)/32)
    const float inv = __expf(-(float)i * 0.28782313662425574f);
    const float ang = (float)l * inv;
    float s, c;
    __sincosf(ang, &s, &c);
    const size_t base = ((size_t)g << 6) + (i << 1);
    v2f qv = *(v2f*)(q + base);
    v2f kv = *(v2f*)(k + base);
    v2f qo = { qv.x * c - qv.y * s, qv.y * c + qv.x * s };
    v2f ko = { kv.x * c - kv.y * s, kv.y * c + kv.x * s };
    *(v2f*)(q + base) = qo;
    *(v2f*)(k + base) = ko;
}

// ---------------------------------------------------------------------------
// Kernel 3: S = q @ k^T / 8 via WMMA, broadcast 9 ENT planes with row/col
// bias + token masks + causal mask. HBM-store bound (302 MB) -> NT stores.
// Block = 8 waves over a 32(m) x 64(n) region; wave = one 16x16 tile.
// ---------------------------------------------------------------------------
__global__ __launch_bounds__(256) void logits_kernel(
    const float* __restrict__ qw, const float* __restrict__ kw,
    const float* __restrict__ rb, const float* __restrict__ cb,
    const int*   __restrict__ tok, float* __restrict__ out)
{
    const int wave = threadIdx.x >> 5;
    const int lane = threadIdx.x & 31;
    const int li   = lane & 15;
    const int hi   = lane >> 4;
    const int b    = blockIdx.z;
    const int m0   = blockIdx.y * 32 + (wave >> 2) * 16;
    const int n0   = blockIdx.x * 64 + (wave & 3) * 16;

    const float* qrow = qw + ((size_t)(b * LL + m0 + li) << 6) + 2 * hi;
    const float* krow = kw + ((size_t)(b * LL + n0 + li) << 6) + 2 * hi;

    v8f acc = {};
    #pragma unroll
    for (int k0 = 0; k0 < DD; k0 += 4) {
        v2f a = *(const v2f*)(qrow + k0);     // A = Q tile 16x4
        v2f f = *(const v2f*)(krow + k0);     // B = K^T tile 4x16
        acc = __builtin_amdgcn_wmma_f32_16x16x4_f32(
                  false, a, false, f, (short)0, acc, false, false);
    }

    const int n  = n0 + li;
    const int cm = tok[b * LL + n];
    int rm[8];
    #pragma unroll
    for (int r = 0; r < 8; ++r) rm[r] = tok[b * LL + m0 + hi * 8 + r];

    #pragma unroll
    for (int e = 0; e < NENT; ++e) {
        const float  cbv = cb[((size_t)(b * NENT + e) << 10) + n];
        const float* rbp = rb + ((size_t)(b * NENT + e) << 10) + m0 + hi * 8;
        float* obase = out + ((size_t)(b * NENT + e) << 20) + n;
        #pragma unroll
        for (int r = 0; r < 8; ++r) {
            const int m = m0 + hi * 8 + r;
            float v = acc[r] * 0.125f + rbp[r] + cbv;
            v = (rm[r] && cm) ? v : -NEPS;   // net effect of the two mask steps
            if (m > n) v -= NEPS;            // strict lower triangle
            // write-once stream: non-temporal, keep L2 for q/k/bias
            __builtin_nontemporal_store(v, obase + ((size_t)m << 10));
        }
    }
}

extern "C" void kernel_launch(void* const* d_in, const int* in_sizes, int n_in,
                              void* d_out, int out_size, void* d_ws, size_t ws_size,
                              hipStream_t stream) {
    const float* lhs = (const float*)d_in[0];
    const float* W1  = (const float*)d_in[1];
    const float* b1  = (const float*)d_in[2];
    const float* W2  = (const float*)d_in[3];
    const float* b2  = (const float*)d_in[4];
    const int*   tok = (const int*)d_in[5];
    float* out = (float*)d_out;

    // Workspace (floats): q(512K) | k(512K) | rb(72K) | cb(72K) | Wct(160K)
    float* qw  = (float*)d_ws;
    float* kw  = qw  + (size_t)BB * LL * DD;
    float* rb  = kw  + (size_t)BB * LL * DD;
    float* cb  = rb  + (size_t)BB * NENT * LL;
    float* Wct = cb  + (size_t)BB * NENT * LL;

    prep_kernel<<<(NCOL * HH) / 256, 256, 0, stream>>>(W1, W2, Wct);
    proj_kernel<<<(512 * (NCOL / 16)) / 8, 256, 0, stream>>>(
        lhs, Wct, b1, b2, qw, kw, rb, cb);
    rope_kernel<<<(BB * LL * 32) / 256, 256, 0, stream>>>(qw, kw);
    logits_kernel<<<dim3(16, 32, 8), 256, 0, stream>>>(qw, kw, rb, cb, tok, out);
}